// GaussianSmoother_30580167147915
// MI455X (gfx1250) — compile-verified
//
#include <hip/hip_runtime.h>
#include <hip/hip_bf16.h>

// D = A x B + C,  V_WMMA_F32_16X16X4_F32 operand types:
typedef __attribute__((ext_vector_type(2))) float v2f;   // A/B: 16x4 / 4x16 f32 = 2 VGPRs
typedef __attribute__((ext_vector_type(8))) float v8f;   // C/D: 16x16 f32 = 8 VGPRs

#define B_BATCH 64
#define T_FULL  2048
#define N_NEUR  1024
#define SIGMA   20.0f
// Truncated Gaussian window: +/- 256 = 12.8 sigma. Tail mass ~exp(-82) ~ 1e-36,
// numerically invisible in fp32. Cuts HBM traffic 512MiB -> 128MiB (4x).
#define WIN     512
#define T0      (T_FULL / 2 - WIN / 2)   // 768
#define WAVES_PER_BLOCK 8

__global__ __launch_bounds__(256) void GaussianSmoother_wmma(
    const float* __restrict__ S, float* __restrict__ out)
{
    __shared__ float kbuf[WIN];
    __shared__ float Zsh;

    const int tid = threadIdx.x;

    // ---- Build normalized Gaussian weights in LDS (deterministic order) ----
    for (int i = tid; i < WIN; i += 256) {
        float x = (float)(i - WIN / 2);            // == t - center, t = T0 + i
        kbuf[i] = expf(-0.5f * (x / SIGMA) * (x / SIGMA));
    }
    __syncthreads();
    if (tid == 0) {                                 // serial sum: bit-deterministic
        float z = 0.0f;
        for (int i = 0; i < WIN; ++i) z += kbuf[i];
        Zsh = z;
    }
    __syncthreads();
    {
        float invZ = 1.0f / Zsh;
        for (int i = tid; i < WIN; i += 256) kbuf[i] *= invZ;
    }
    __syncthreads();

    // ---- Tile assignment: one wave -> one (batch, 16-neuron) output tile ----
    const int wave = tid >> 5;
    const int lane = tid & 31;
    const int tile = blockIdx.x * WAVES_PER_BLOCK + wave;   // 0..4095
    const int b    = tile >> 6;                             // 64 n-tiles per batch
    const int n0   = (tile & 63) << 4;
    const int nIdx = lane & 15;
    const int kOff = (lane >> 4) << 1;                      // lanes 0-15 -> K{0,1}, 16-31 -> K{2,3}

    // Per ISA 16x4 f32 A layout: VGPR0 = K=0|K=2, VGPR1 = K=1|K=3 across lane halves.
    // B (4x16) mirrored: lane half selects K pair, lane&15 selects N column.
    const float* p = S + ((size_t)b * T_FULL + (size_t)(T0 + kOff)) * N_NEUR
                       + (size_t)(n0 + nIdx);

    v8f acc = {};

#pragma unroll 4
    for (int i = 0; i < WIN; i += 4) {
        // Spike tile: two dword loads, 4KB apart (consecutive time steps).
        // Each half-wave touches one contiguous 64B segment per load.
        float s0 = p[0];
        float s1 = p[N_NEUR];
        v2f bm;
        bm.x = s0;                                  // B[K=0|2, n]
        bm.y = s1;                                  // B[K=1|3, n]

        // Kernel weights broadcast down M: A[m,k] = k[t0+i+k] for all m.
        v2f am = *(const v2f*)&kbuf[i + kOff];      // ds_load_b64 (8B aligned: i%4==0, kOff even)

        acc = __builtin_amdgcn_wmma_f32_16x16x4_f32(
            /*neg_a=*/false, am, /*neg_b=*/false, bm,
            /*c_mod=*/(short)0, acc, /*reuse_a=*/false, /*reuse_b=*/false);

        p += 4 * N_NEUR;
    }

    // D rows are all identical (A rows identical); row M=0 lives in VGPR0, lanes 0-15.
    if (lane < 16) {
        out[(size_t)b * N_NEUR + (size_t)(n0 + lane)] = acc[0];
    }
}

extern "C" void kernel_launch(void* const* d_in, const int* in_sizes, int n_in,
                              void* d_out, int out_size, void* d_ws, size_t ws_size,
                              hipStream_t stream) {
    const float* S   = (const float*)d_in[0];     // [64, 2048, 1024] fp32
    float*       out = (float*)d_out;             // [64, 1024] fp32

    // 64 batches * 64 n-tiles = 4096 waves; 8 waves per 256-thread block.
    const int total_tiles = B_BATCH * (N_NEUR / 16);
    const int blocks = total_tiles / WAVES_PER_BLOCK;       // 512
    GaussianSmoother_wmma<<<blocks, 256, 0, stream>>>(S, out);
}